// Layer_91164975825062
// MI455X (gfx1250) — compile-verified
//
#include <hip/hip_runtime.h>

typedef __attribute__((ext_vector_type(2))) float v2f;
typedef __attribute__((ext_vector_type(8))) float v8f;

#define FD 128  // feature dim

// ---------------- zero the output accumulator ----------------
__global__ void zero_kernel(float4* __restrict__ z, int n4) {
  int i = blockIdx.x * blockDim.x + threadIdx.x;
  if (i < n4) z[i] = make_float4(0.f, 0.f, 0.f, 0.f);
}

// ---------------- per-node gate GEMV via f32 WMMA ----------------
// g[i*2+0] = h[i] . gate_w[0:128]   (w_d)
// g[i*2+1] = h[i] . gate_w[128:256] (w_s)
// One wave handles a 16-node tile using V_WMMA_F32_16X16X4_F32, chaining
// 32 WMMAs over K=128. B column 0 carries w_d, column 1 carries w_s,
// columns 2..15 are zeroed via a loop-invariant per-lane mask (branchless).
__global__ void gate_kernel(const float* __restrict__ h,
                            const float* __restrict__ gw,
                            float* __restrict__ g, int N) {
  int wave = (int)((blockIdx.x * (unsigned)blockDim.x + threadIdx.x) >> 5);
  int lane = (int)(threadIdx.x & 31u);
  int r0 = wave * 16;
  if (r0 >= N) return;  // wave-uniform: EXEC stays all-ones for WMMA

  // A layout (32-bit A 16x4): lanes 0-15 -> M=lane, hold K={0,1};
  //                           lanes 16-31 -> M=lane-16, hold K={2,3}
  int mrow  = lane & 15;
  int khalf = (lane >> 4) << 1;          // 0 or 2
  // B layout (4x16): lane holds column n = lane&15; VGPR0 -> K=khalf, VGPR1 -> K=khalf+1
  int n = lane & 15;

  const float* hrow = h + (size_t)(r0 + mrow) * FD;
  // Branchless B: address select (w_d vs w_s) + zero-mask for columns >= 2.
  const float* wrow  = gw + ((n == 1) ? FD : 0);
  float        scale = (n < 2) ? 1.f : 0.f;

  v8f c = {0.f, 0.f, 0.f, 0.f, 0.f, 0.f, 0.f, 0.f};
  #pragma unroll 4
  for (int kc = 0; kc < 32; ++kc) {
    int k = kc * 4 + khalf;
    v2f a;
    a.x = hrow[k];
    a.y = hrow[k + 1];
    v2f b;
    b.x = wrow[k]     * scale;
    b.y = wrow[k + 1] * scale;
    c = __builtin_amdgcn_wmma_f32_16x16x4_f32(false, a, false, b,
                                              (short)0, c, false, false);
  }

  // C layout: VGPR r, lanes 0-15 -> (M=r, N=lane), lanes 16-31 -> (M=r+8)
  if (n < 2) {
    int mbase = (lane >> 4) * 8;
    #pragma unroll
    for (int r = 0; r < 8; ++r)
      g[(size_t)(r0 + mbase + r) * 2 + n] = c[r];
  }
}

// ---------------- per-edge gather + gated scatter-add ----------------
// One wave per edge: lane j handles features [4j, 4j+4).
__global__ void edge_kernel(const float* __restrict__ h,
                            const float* __restrict__ deg,
                            const float* __restrict__ g,
                            const float* __restrict__ gate_b,
                            const int* __restrict__ src,
                            const int* __restrict__ dst,
                            float* __restrict__ z, int E) {
  int wave = (int)((blockIdx.x * (unsigned)blockDim.x + threadIdx.x) >> 5);
  int lane = (int)(threadIdx.x & 31u);
  if (wave >= E) return;

  int s = src[wave];
  int d = dst[wave];

  float a  = tanhf(g[2 * (size_t)d] + g[2 * (size_t)s + 1] + gate_b[0]);
  float ev = a * deg[d] * deg[s];

  const float4* hs = (const float4*)(h + (size_t)s * FD);
  float4 v = hs[lane];                       // coalesced 512B row gather
  float* zr = z + (size_t)d * FD + lane * 4;
  unsafeAtomicAdd(zr + 0, v.x * ev);         // -> global_atomic_add_f32
  unsafeAtomicAdd(zr + 1, v.y * ev);
  unsafeAtomicAdd(zr + 2, v.z * ev);
  unsafeAtomicAdd(zr + 3, v.w * ev);
}

extern "C" void kernel_launch(void* const* d_in, const int* in_sizes, int n_in,
                              void* d_out, int out_size, void* d_ws, size_t ws_size,
                              hipStream_t stream) {
  const float* h      = (const float*)d_in[0];   // (N, 128) f32
  const float* deg    = (const float*)d_in[1];   // (N,)     f32
  const float* gate_w = (const float*)d_in[2];   // (1, 256) f32
  const float* gate_b = (const float*)d_in[3];   // (1,)     f32
  const int*   src    = (const int*)d_in[4];     // (E,)
  const int*   dst    = (const int*)d_in[5];     // (E,)
  float* z = (float*)d_out;                      // (N, 128) f32

  int N = in_sizes[0] / FD;
  int E = in_sizes[4];

  float* g = (float*)d_ws;                       // N*2 floats of scratch

  // 1) zero z
  int n4 = (N * FD) / 4;
  zero_kernel<<<(n4 + 255) / 256, 256, 0, stream>>>((float4*)z, n4);

  // 2) gate GEMV: one wave per 16 nodes
  int waves = (N + 15) / 16;
  int threads = waves * 32;
  gate_kernel<<<(threads + 255) / 256, 256, 0, stream>>>(h, gate_w, g, N);

  // 3) edge scatter: one wave per edge
  long long ethreads = (long long)E * 32;
  int eblocks = (int)((ethreads + 255) / 256);
  edge_kernel<<<eblocks, 256, 0, stream>>>(h, deg, g, gate_b, src, dst, z, E);
}